// GNNModel_37005438222986
// MI455X (gfx1250) — compile-verified
//
#include <hip/hip_runtime.h>
#include <cstdint>
#include <cstddef>

#define EPSV 1e-5f
#define DH 128

typedef __attribute__((ext_vector_type(2))) float v2f;
typedef __attribute__((ext_vector_type(8))) float v8f;

// ---------------- degree / normalization coefficients ----------------
__global__ void deg_init_kernel(float* __restrict__ deg, int N) {
    int i = blockIdx.x * blockDim.x + threadIdx.x;
    if (i < N) deg[i] = 1.0f;  // self-loop
}

__global__ void deg_edges_kernel(const long long* __restrict__ dst,
                                 float* __restrict__ deg, int E) {
    int e = blockIdx.x * blockDim.x + threadIdx.x;
    if (e < E) atomicAdd(&deg[(int)dst[e]], 1.0f);
}

__global__ void rsqrt_kernel(float* __restrict__ d, int N) {
    int i = blockIdx.x * blockDim.x + threadIdx.x;
    if (i < N) {
        float v = d[i];
        d[i] = v > 0.0f ? rsqrtf(v) : 0.0f;
    }
}

// ---------------- GEMM: Out(N x 128) = A(N x K) @ W(K x 128) ----------------
// grid.x = N/16 row tiles; 256 threads = 8 waves; wave w -> column tile w.
// Uses V_WMMA_F32_16X16X4_F32. Lane layout per CDNA5 ISA 7.12.2:
//   A 16x4 f32: lanes 0-15 hold M=lane, K={k0,k0+1}; lanes 16-31 K={k0+2,k0+3}
//   B  4x16 f32: mirrored (N=lane%16, same K striping)
//   C/D 16x16:   VGPR j -> M = j + 8*(lane>=16), N = lane%16
__global__ __launch_bounds__(256) void gemm_wmma_kernel(
    const float* __restrict__ A, const float* __restrict__ W,
    float* __restrict__ Out, int K) {
    const int lane = threadIdx.x & 31;
    const int wave = threadIdx.x >> 5;
    const int half = lane >> 4;     // 0 or 1
    const int l16  = lane & 15;
    const int rowBase = blockIdx.x << 4;
    const int col = (wave << 4) + l16;
    const float* arow = A + (size_t)(rowBase + l16) * K;

    v8f acc = {};
    for (int k0 = 0; k0 < K; k0 += 4) {
        const int kk = k0 + (half << 1);
        v2f a, b;
        a.x = arow[kk];
        a.y = arow[kk + 1];
        b.x = W[(size_t)kk * DH + col];
        b.y = W[(size_t)(kk + 1) * DH + col];
        acc = __builtin_amdgcn_wmma_f32_16x16x4_f32(
            false, a, false, b, (short)0, acc, false, false);
    }
#pragma unroll
    for (int j = 0; j < 8; ++j)
        Out[(size_t)(rowBase + j + (half << 3)) * DH + col] = acc[j];
}

// ---------------- GEMV: out(N) = A(N x 128) @ w(128) ----------------
__global__ void gemv128_kernel(const float* __restrict__ A, const float* __restrict__ w,
                               float* __restrict__ out, int N) {
    int node = blockIdx.x * (blockDim.x >> 5) + (threadIdx.x >> 5);
    int lane = threadIdx.x & 31;
    if (node >= N) return;
    const float* row = A + (size_t)node * DH;
    float s = 0.0f;
#pragma unroll
    for (int k = 0; k < 4; ++k) s += row[lane + 32 * k] * w[lane + 32 * k];
#pragma unroll
    for (int off = 16; off > 0; off >>= 1) s += __shfl_xor(s, off, 32);
    if (lane == 0) out[node] = s;
}

// ---------------- aggregation: agg = b + D^-1/2 (A+I) D^-1/2 h ----------------
__global__ void agg_init_kernel(const float* __restrict__ h, const float* __restrict__ bias,
                                const float* __restrict__ dinv, float* __restrict__ agg,
                                int N, int C) {
    int i = blockIdx.x * blockDim.x + threadIdx.x;
    if (i >= N * C) return;
    int node = i / C, c = i - node * C;
    float di = dinv[node];
    agg[i] = bias[c] + h[i] * di * di;   // bias + self-loop term
}

// one block (128 lanes) per edge, channel = lane: coalesced gather + atomics (L2-resident)
__global__ void agg_edges128_kernel(const float* __restrict__ h,
                                    const long long* __restrict__ src,
                                    const long long* __restrict__ dst,
                                    const float* __restrict__ dinv,
                                    float* __restrict__ agg, int E) {
    int e = blockIdx.x;
    if (e >= E) return;
    int s = (int)src[e], d = (int)dst[e];
    float nrm = dinv[s] * dinv[d];
    int c = threadIdx.x;
    atomicAdd(&agg[(size_t)d * DH + c], h[(size_t)s * DH + c] * nrm);
}

__global__ void agg_edges1_kernel(const float* __restrict__ h,
                                  const long long* __restrict__ src,
                                  const long long* __restrict__ dst,
                                  const float* __restrict__ dinv,
                                  float* __restrict__ agg, int E) {
    int e = blockIdx.x * blockDim.x + threadIdx.x;
    if (e >= E) return;
    int s = (int)src[e], d = (int)dst[e];
    atomicAdd(&agg[d], h[s] * dinv[s] * dinv[d]);
}

// ---------------- norm tower (BN -> InstanceNorm -> graph LayerNorm) ----------------
__global__ void zero_kernel(float* __restrict__ p, int n) {
    int i = blockIdx.x * blockDim.x + threadIdx.x;
    if (i < n) p[i] = 0.0f;
}

// per-column sum / sumsq; blockDim fixed at 128, generic over C (1 or 128)
__global__ void col_stats_kernel(const float* __restrict__ h, float* __restrict__ sums,
                                 int N, int C, int rowsPerBlock) {
    int c = threadIdx.x % C;
    int rstep = blockDim.x / C;
    int r0 = blockIdx.x * rowsPerBlock + threadIdx.x / C;
    int rend = blockIdx.x * rowsPerBlock + rowsPerBlock;
    if (rend > N) rend = N;
    float s = 0.0f, sq = 0.0f;
    for (int r = r0; r < rend; r += rstep) {
        float v = h[(size_t)r * C + c];
        s += v;
        sq += v * v;
    }
    atomicAdd(&sums[c], s);
    atomicAdd(&sums[C + c], sq);
}

// Collapse BN->InstanceNorm->LayerNorm(graph) into per-channel affine.
// BN:   y  = (h-m)*r1*g + bt          (column mean bt, var g^2 r1^2 v)
// IN:   y2 = (h-m)*g*r1*rsqrt(g^2 r1^2 v + eps)   (bt cancels; col mean 0, var v3)
// LN:   global mean 0, global var = mean_c(v3) -> final per-channel scale/shift.
__global__ void finalize_norm_kernel(const float* __restrict__ sums,
                                     const float* __restrict__ g,
                                     const float* __restrict__ lw,
                                     const float* __restrict__ lb,
                                     float* __restrict__ scale, float* __restrict__ shift,
                                     int N, int C) {
    __shared__ float sh[128];
    int c = threadIdx.x;
    float v3 = 0.0f, alpha = 0.0f, m = 0.0f;
    if (c < C) {
        m = sums[c] / (float)N;
        float v = sums[C + c] / (float)N - m * m;
        v = fmaxf(v, 0.0f);
        float r1 = rsqrtf(v + EPSV);
        float a1 = g[c] * r1;                 // BN scale
        float vy = a1 * a1 * v;               // BN output column variance
        alpha = a1 * rsqrtf(vy + EPSV);       // BN+IN combined scale
        v3 = alpha * alpha * v;               // IN output column variance
    }
    sh[threadIdx.x] = (c < C) ? v3 : 0.0f;
    __syncthreads();
    for (int off = 64; off > 0; off >>= 1) {
        if (threadIdx.x < off) sh[threadIdx.x] += sh[threadIdx.x + off];
        __syncthreads();
    }
    float Vtot = sh[0] / (float)C;            // LayerNorm global variance
    float rln = rsqrtf(Vtot + EPSV);
    if (c < C) {
        float sc = alpha * rln * lw[c];
        scale[c] = sc;
        shift[c] = lb[c] - m * sc;
    }
}

__global__ void apply_affine_kernel(const float* __restrict__ h,
                                    const float* __restrict__ scale,
                                    const float* __restrict__ shift,
                                    float* __restrict__ out, int total, int C, int relu) {
    int i = blockIdx.x * blockDim.x + threadIdx.x;
    if (i >= total) return;
    int c = i % C;
    float v = h[i] * scale[c] + shift[c];
    if (relu) v = fmaxf(v, 0.0f);
    out[i] = v;
}

// ---------------- launch ----------------
extern "C" void kernel_launch(void* const* d_in, const int* in_sizes, int n_in,
                              void* d_out, int out_size, void* d_ws, size_t ws_size,
                              hipStream_t stream) {
    const float* x  = (const float*)d_in[0];
    const long long* ei = (const long long*)d_in[1];   // int64 (2, E)
    const float* W1 = (const float*)d_in[2];
    const float* b1 = (const float*)d_in[3];
    const float* g1 = (const float*)d_in[4];
    const float* lw1 = (const float*)d_in[6];
    const float* lb1 = (const float*)d_in[7];
    const float* W2 = (const float*)d_in[8];
    const float* b2 = (const float*)d_in[9];
    const float* g2 = (const float*)d_in[10];
    const float* lw2 = (const float*)d_in[12];
    const float* lb2 = (const float*)d_in[13];
    const float* W3 = (const float*)d_in[14];
    const float* b3 = (const float*)d_in[15];
    const float* g3 = (const float*)d_in[16];
    const float* lw3 = (const float*)d_in[18];
    const float* lb3 = (const float*)d_in[19];

    const int N = in_sizes[0] / 32;
    const int E = in_sizes[1] / 2;
    const long long* srcI = ei;
    const long long* dstI = ei + E;

    float* ws = (float*)d_ws;
    float* H0    = ws;                       // N x 128
    float* H1    = H0 + (size_t)N * DH;      // N x 128
    float* dinv  = H1 + (size_t)N * DH;      // N
    float* colA  = dinv + N;                 // N
    float* colB  = colA + N;                 // N
    float* stats = colB + N;                 // 256
    float* scl   = stats + 256;              // 128
    float* shf   = scl + 128;                // 128

    const int T = 256;
    const int gN   = (N + T - 1) / T;
    const int gE   = (E + T - 1) / T;
    const int gNC  = (N * DH + T - 1) / T;
    const int ROWS = 512;
    const int gST  = (N + ROWS - 1) / ROWS;
    const int gTile = N / 16;                // N = 100000 -> 6250 tiles

    // degree -> dinv
    deg_init_kernel<<<gN, T, 0, stream>>>(dinv, N);
    deg_edges_kernel<<<gE, T, 0, stream>>>(dstI, dinv, E);
    rsqrt_kernel<<<gN, T, 0, stream>>>(dinv, N);

    // ---- layer 1: gemm x -> H0 ; agg H0 -> H1 ; norms+relu H1 -> H0 ----
    gemm_wmma_kernel<<<gTile, 256, 0, stream>>>(x, W1, H0, 32);
    agg_init_kernel<<<gNC, T, 0, stream>>>(H0, b1, dinv, H1, N, DH);
    agg_edges128_kernel<<<E, DH, 0, stream>>>(H0, srcI, dstI, dinv, H1, E);
    zero_kernel<<<1, 256, 0, stream>>>(stats, 256);
    col_stats_kernel<<<gST, 128, 0, stream>>>(H1, stats, N, DH, ROWS);
    finalize_norm_kernel<<<1, 128, 0, stream>>>(stats, g1, lw1, lb1, scl, shf, N, DH);
    apply_affine_kernel<<<gNC, T, 0, stream>>>(H1, scl, shf, H0, N * DH, DH, 1);

    // ---- layer 2: gemm H0 -> H1 ; agg H1 -> H0 ; norms+relu H0 -> H1 ----
    gemm_wmma_kernel<<<gTile, 256, 0, stream>>>(H0, W2, H1, DH);
    agg_init_kernel<<<gNC, T, 0, stream>>>(H1, b2, dinv, H0, N, DH);
    agg_edges128_kernel<<<E, DH, 0, stream>>>(H1, srcI, dstI, dinv, H0, E);
    zero_kernel<<<1, 256, 0, stream>>>(stats, 256);
    col_stats_kernel<<<gST, 128, 0, stream>>>(H0, stats, N, DH, ROWS);
    finalize_norm_kernel<<<1, 128, 0, stream>>>(stats, g2, lw2, lb2, scl, shf, N, DH);
    apply_affine_kernel<<<gNC, T, 0, stream>>>(H0, scl, shf, H1, N * DH, DH, 1);

    // ---- layer 3 (C=1): gemv H1 -> colA ; agg colA -> colB ; norms -> d_out ----
    gemv128_kernel<<<(N + 7) / 8, 256, 0, stream>>>(H1, W3, colA, N);
    agg_init_kernel<<<gN, T, 0, stream>>>(colA, b3, dinv, colB, N, 1);
    agg_edges1_kernel<<<gE, T, 0, stream>>>(colA, srcI, dstI, dinv, colB, E);
    zero_kernel<<<1, 256, 0, stream>>>(stats, 256);
    col_stats_kernel<<<gST, 128, 0, stream>>>(colB, stats, N, 1, ROWS);
    finalize_norm_kernel<<<1, 128, 0, stream>>>(stats, g3, lw3, lb3, scl, shf, N, 1);
    apply_affine_kernel<<<gN, T, 0, stream>>>(colB, scl, shf, (float*)d_out, N, 1, 0);
}